// ObjectDetector_48498770706448
// MI455X (gfx1250) — compile-verified
//
#include <hip/hip_runtime.h>

#define HH 96
#define WW 96
#define AA 9
#define NN (HH * WW * AA)            // 82944
#define TOPN 2000
#define THRESH 0.5f
#define NMS_THREADS 1024
#define ITEMS (NN / NMS_THREADS)     // 81 exactly

typedef float v2f __attribute__((ext_vector_type(2)));
typedef float v8f __attribute__((ext_vector_type(8)));

// ---------------------------------------------------------------------------
// Kernel 1: sigmoid + bbox decode + clip, and per-box areas via WMMA
// (area = w*h expressed as a 16x16x4 f32 outer product; diagonal extracted).
// Grid is exactly NN threads -> EXEC all ones at the WMMA (required by ISA).
// ---------------------------------------------------------------------------
__global__ __launch_bounds__(256)
void decode_kernel(const float* __restrict__ cls,
                   const float4* __restrict__ deltas,
                   const float4* __restrict__ anchors,
                   float* __restrict__ scores,
                   float4* __restrict__ props,
                   float* __restrict__ areas)
{
    const int i = blockIdx.x * blockDim.x + threadIdx.x;   // < NN exactly

    // sigmoid(cls) -> live score buffer
    const float sc = cls[i];
    scores[i] = 1.0f / (1.0f + expf(-sc));

    // Faster-RCNN decode
    const float4 d = deltas[i];
    const float4 a = anchors[i];
    const float aw  = a.z - a.x;
    const float ah  = a.w - a.y;
    const float acx = a.x + 0.5f * aw;
    const float acy = a.y + 0.5f * ah;
    const float pcx = d.x * aw + acx;
    const float pcy = d.y * ah + acy;
    const float pw  = expf(d.z) * aw;
    const float ph  = expf(d.w) * ah;
    float x1 = pcx - 0.5f * pw;
    float y1 = pcy - 0.5f * ph;
    float x2 = pcx + 0.5f * pw;
    float y2 = pcy + 0.5f * ph;
    x1 = fminf(fmaxf(x1, 0.0f), 1.0f);
    y1 = fminf(fmaxf(y1, 0.0f), 1.0f);
    x2 = fminf(fmaxf(x2, 0.0f), 1.0f);
    y2 = fminf(fmaxf(y2, 0.0f), 1.0f);
    props[i] = make_float4(x1, y1, x2, y2);

    const float w = x2 - x1;
    const float h = y2 - y1;

#if defined(__gfx1250__) && __has_builtin(__builtin_amdgcn_wmma_f32_16x16x4_f32)
    // Areas for the wave's 32 boxes via two 16x16x4 f32 WMMAs.
    // A[m][0]=w_m (lanes 0-15 hold K=0 in VGPR0), B[0][n]=h_n, all other K zero:
    // D[m][n] = w_m * h_n + 0, diagonal D[m][m] = exact fp32 w_m*h_m.
    const int lane = threadIdx.x & 31;
    const int waveBase = i - lane;
    const float whi = __shfl_down(w, 16, 32);   // lanes 0-15 get boxes 16-31
    const float hhi = __shfl_down(h, 16, 32);
#pragma unroll
    for (int j = 0; j < 2; ++j) {
        const float wj = j ? whi : w;
        const float hj = j ? hhi : h;
        v2f av; av.x = (lane < 16) ? wj : 0.0f; av.y = 0.0f;
        v2f bv; bv.x = (lane < 16) ? hj : 0.0f; bv.y = 0.0f;
        v8f c = {};
        v8f dm = __builtin_amdgcn_wmma_f32_16x16x4_f32(
            false, av, false, bv, (short)0, c, false, false);
        // Diagonal: box m<8 -> VGPR m, lane m ; box m>=8 -> VGPR m-8, lane m+16.
        union { v8f v; float f[8]; } u; u.v = dm;
        const float diag = u.f[lane & 7];
        if (lane < 8)
            areas[waveBase + j * 16 + lane] = diag;          // boxes 0..7 of tile
        else if (lane >= 24)
            areas[waveBase + j * 16 + (lane - 16)] = diag;   // boxes 8..15 of tile
    }
#else
    areas[i] = w * h;
#endif
}

// ---------------------------------------------------------------------------
// Kernel 2: persistent single-workgroup soft-NMS (2000 sequential steps).
// 1024 threads = 32 wave32 waves on one WGP; each thread owns gid = t + k*1024.
// Argmax ties resolved to lowest index (== jnp.argmax first occurrence).
// ---------------------------------------------------------------------------
__global__ __launch_bounds__(NMS_THREADS, 1)
void softnms_kernel(const float4* __restrict__ props,
                    const float* __restrict__ areas,
                    float* __restrict__ scores,
                    float* __restrict__ out_boxes,   // TOPN*4
                    float* __restrict__ out_scores)  // TOPN
{
    __shared__ float red_val[32];
    __shared__ int   red_idx[32];
    __shared__ float bc_box[4];
    __shared__ float bc_a1;
    __shared__ int   bc_idx;
    __shared__ int   bc_valid;

    const int t    = threadIdx.x;
    const int lane = t & 31;
    const int wid  = t >> 5;

    for (int iter = 0; iter < TOPN; ++iter) {
        // ---------- Phase A: global argmax ----------
        float best = -1.0f;
        int   bgid = 0;
#pragma unroll 3
        for (int k = 0; k < ITEMS; ++k) {
            const int gid = t + (k << 10);
            const float s = scores[gid];
            if (s > best) { best = s; bgid = gid; }   // ascending gid -> first max
        }
#pragma unroll
        for (int off = 16; off > 0; off >>= 1) {
            const float ov = __shfl_xor(best, off, 32);
            const int   og = __shfl_xor(bgid, off, 32);
            if (ov > best || (ov == best && og < bgid)) { best = ov; bgid = og; }
        }
        if (lane == 0) { red_val[wid] = best; red_idx[wid] = bgid; }
        __syncthreads();
        if (wid == 0) {
            best = red_val[lane];
            bgid = red_idx[lane];
#pragma unroll
            for (int off = 16; off > 0; off >>= 1) {
                const float ov = __shfl_xor(best, off, 32);
                const int   og = __shfl_xor(bgid, off, 32);
                if (ov > best || (ov == best && og < bgid)) { best = ov; bgid = og; }
            }
            if (lane == 0) {
                const int valid = (best > THRESH) ? 1 : 0;
                const float4 bb = props[bgid];
                out_boxes[4 * iter + 0] = valid ? bb.x : 0.0f;
                out_boxes[4 * iter + 1] = valid ? bb.y : 0.0f;
                out_boxes[4 * iter + 2] = valid ? bb.z : 0.0f;
                out_boxes[4 * iter + 3] = valid ? bb.w : 0.0f;
                out_scores[iter]        = valid ? best : 0.0f;
                bc_box[0] = bb.x; bc_box[1] = bb.y;
                bc_box[2] = bb.z; bc_box[3] = bb.w;
                bc_a1  = areas[bgid];          // WMMA-precomputed area
                bc_idx = bgid;
                bc_valid = valid;
            }
        }
        __syncthreads();

        if (!bc_valid) {
            // Max score can only decrease; every remaining output is exactly 0.
            for (int j = iter + 1 + t; j < TOPN; j += NMS_THREADS) {
                out_scores[j] = 0.0f;
                out_boxes[4 * j + 0] = 0.0f;
                out_boxes[4 * j + 1] = 0.0f;
                out_boxes[4 * j + 2] = 0.0f;
                out_boxes[4 * j + 3] = 0.0f;
            }
            return;   // uniform across the block
        }

        const float bx1 = bc_box[0], by1 = bc_box[1];
        const float bx2 = bc_box[2], by2 = bc_box[3];
        const float a1  = bc_a1;
        const int   topgid = bc_idx;

        // ---------- Phase B: soft decay ----------
#pragma unroll 3
        for (int k = 0; k < ITEMS; ++k) {
            const int gid = t + (k << 10);
            const float  s = scores[gid];
            const float4 p = props[gid];
            const float a2    = (p.z - p.x) * (p.w - p.y);
            const float ix1   = fmaxf(bx1, p.x);
            const float iy1   = fmaxf(by1, p.y);
            const float ix2   = fminf(bx2, p.z);
            const float iy2   = fminf(by2, p.w);
            const float inter = fmaxf(ix2 - ix1, 0.0f) * fmaxf(iy2 - iy1, 0.0f);
            const float uni   = a1 + a2 - inter;
            const float iou   = (uni > 0.0f) ? (inter / fmaxf(uni, 1e-12f)) : 0.0f;
            const float sn    = (gid == topgid) ? 0.0f : s * expf(-iou * iou);
            scores[gid] = sn;
        }
        __syncthreads();
    }
}

// ---------------------------------------------------------------------------
// inputs: [0]=rpn_cls_score (82944 f32), [1]=rpn_bbox_pred (331776 f32),
//         [2]=anchors (331776 f32)
// d_out:  2000*4 boxes then 2000 scores (f32), flat.
// d_ws:   scores[NN] | areas[NN] | props[NN] (float4)  ~1.9 MB
// ---------------------------------------------------------------------------
extern "C" void kernel_launch(void* const* d_in, const int* in_sizes, int n_in,
                              void* d_out, int out_size, void* d_ws, size_t ws_size,
                              hipStream_t stream)
{
    (void)in_sizes; (void)n_in; (void)out_size; (void)ws_size;
    const float* cls     = (const float*)d_in[0];
    const float* deltas  = (const float*)d_in[1];
    const float* anchors = (const float*)d_in[2];
    float* out = (float*)d_out;

    char* ws = (char*)d_ws;
    float*  scores = (float*)ws;                                  // NN floats
    float*  areas  = (float*)(ws + (size_t)NN * sizeof(float));   // NN floats
    float4* props  = (float4*)(ws + (size_t)2 * NN * sizeof(float)); // 16B aligned

    decode_kernel<<<NN / 256, 256, 0, stream>>>(
        cls, (const float4*)deltas, (const float4*)anchors, scores, props, areas);

    softnms_kernel<<<1, NMS_THREADS, 0, stream>>>(
        props, areas, scores, out, out + 4 * TOPN);
}